// Bi_Mamba_2199023255672
// MI455X (gfx1250) — compile-verified
//
#include <hip/hip_runtime.h>
#include <hip/hip_bf16.h>
#include <math.h>

// ---------------------------------------------------------------------------
// Bi-directional Mamba block for gfx1250 (MI455X).
// fp32 end-to-end; GEMMs use V_WMMA_F32_16X16X4_F32 (wave32 matrix pipe) with
// double-buffered GLOBAL_LOAD_ASYNC_TO_LDS staging (ASYNCcnt pipeline) and
// two N-adjacent accumulators per wave (16x32 per wave, 64x64 block tile).
// Sequence flips are folded into reversed-time conv + scan (flip commutes
// with all position-wise ops), saving two full-tensor flips of HBM traffic.
// ---------------------------------------------------------------------------

#define D_MODEL   384
#define D_INNER   768
#define D_STATE   16
#define DT_RANK   24
#define XPROJ_N   (DT_RANK + 2 * D_STATE)   // 56
#define K_CONV    4
#define BATCH     8
#define SEQ       2048
#define NTOK      (BATCH * SEQ)             // 16384

typedef __attribute__((ext_vector_type(2))) float v2f;
typedef __attribute__((ext_vector_type(8))) float v8f;

__device__ __forceinline__ float silu_f(float v) {
    return v / (1.0f + __expf(-v));
}

// LDS byte offset of a __shared__ object (generic-pointer low 32 bits are the
// wave-relative LDS address per the CDNA5 aperture mapping).
__device__ __forceinline__ unsigned lds_addr32(const void* p) {
    return (unsigned)(unsigned long long)p;
}

// Async global->LDS copy (tracked by ASYNCcnt).
__device__ __forceinline__ void async_to_lds_b128(unsigned lds, const float* g) {
    asm volatile("global_load_async_to_lds_b128 %0, %1, off"
                 :: "v"(lds), "v"(g) : "memory");
}

// ---------------------------------------------------------------------------
// LayerNorm: one wave32 per row of 384 (12 elements / lane).
// ---------------------------------------------------------------------------
__global__ __launch_bounds__(128)
void ln_kernel(const float* __restrict__ in, const float* __restrict__ w,
               const float* __restrict__ b, float* __restrict__ out) {
    int row  = blockIdx.x * 4 + (threadIdx.x >> 5);
    int lane = threadIdx.x & 31;
    const float* r = in + (size_t)row * D_MODEL;

    float v[12];
    float s = 0.0f;
#pragma unroll
    for (int i = 0; i < 12; ++i) { v[i] = r[lane + 32 * i]; s += v[i]; }
#pragma unroll
    for (int off = 16; off > 0; off >>= 1) s += __shfl_xor(s, off, 32);
    float mu = s * (1.0f / D_MODEL);

    float var = 0.0f;
#pragma unroll
    for (int i = 0; i < 12; ++i) { float d = v[i] - mu; var += d * d; }
#pragma unroll
    for (int off = 16; off > 0; off >>= 1) var += __shfl_xor(var, off, 32);
    float rstd = rsqrtf(var * (1.0f / D_MODEL) + 1e-5f);

    float* o = out + (size_t)row * D_MODEL;
#pragma unroll
    for (int i = 0; i < 12; ++i) {
        int c = lane + 32 * i;
        o[c] = (v[i] - mu) * rstd * w[c] + b[c];
    }
}

// ---------------------------------------------------------------------------
// WMMA GEMM:  C[M,N] = A[M,K] * W[N,K]^T  (+ optional residual R[M,N]).
// 256 threads = 8 waves. Block tile 64(M) x 64(N); each wave owns a 16x32
// strip (2 accumulators, A-fragment reused across both WMMAs per K-step).
// K staged in chunks of 16 through double-buffered LDS filled with
// global_load_async_to_lds_b128: exactly 2 async ops per wave per tile
// (one for A, one for B), so s_wait_asynccnt 0x2 overlaps tile i+1 fill
// with tile i compute. Fragment layouts per CDNA5 ISA 7.12.2.
// Rows padded to 20 floats: 16B-aligned b128 stores and conflict-free
// 16-lane fragment reads (gcd(20,64)=4 -> 16 distinct banks).
// ---------------------------------------------------------------------------
__global__ __launch_bounds__(256)
void gemm_wmma_kernel(const float* __restrict__ A, const float* __restrict__ W,
                      const float* __restrict__ R, float* __restrict__ C,
                      int M, int N, int K) {
    __shared__ __align__(16) float As[2][64][20];   // [buf][m][k]
    __shared__ __align__(16) float Bs[2][64][20];   // [buf][n][k]

    int tid  = threadIdx.x;
    int wave = tid >> 5;
    int lane = tid & 31;
    int wm   = wave & 3;          // 0..3 : M sub-tile (16 rows)
    int wn   = wave >> 2;         // 0..1 : N strip (32 cols = 2 tiles)
    int half = lane >> 4;         // lane group (0: lanes 0-15, 1: 16-31)
    int l    = lane & 15;

    size_t bm = (size_t)blockIdx.y * 64;
    int    bn = blockIdx.x * 64;

    // staging map: A tile 64x16 and B tile 64x16 -> one b128 per thread each.
    int srr = tid >> 2, sc4 = (tid & 3) * 4;       // staged row, col
    const float* gA = A + (bm + srr) * (size_t)K + sc4;
    const float* gB = W + (size_t)(bn + srr) * K + sc4;

    v8f acc0 = {};
    v8f acc1 = {};

    const int NT = K / 16;
    // prologue: tile 0 -> buffer 0
    async_to_lds_b128(lds_addr32(&As[0][srr][sc4]), gA);
    async_to_lds_b128(lds_addr32(&Bs[0][srr][sc4]), gB);

    for (int it = 0; it < NT; ++it) {
        int buf = it & 1;
        bool has_next = (it + 1 < NT);
        if (has_next) {                    // fill other buffer ahead
            int k0n = (it + 1) * 16;
            async_to_lds_b128(lds_addr32(&As[buf ^ 1][srr][sc4]), gA + k0n);
            async_to_lds_b128(lds_addr32(&Bs[buf ^ 1][srr][sc4]), gB + k0n);
            asm volatile("s_wait_asynccnt 0x2" ::: "memory");
        } else {
            asm volatile("s_wait_asynccnt 0x0" ::: "memory");
        }
        __syncthreads();                   // tile `it` visible to all waves

#pragma unroll
        for (int ks = 0; ks < 16; ks += 4) {
            int klo = ks + (half ? 2 : 0);
            int khi = ks + (half ? 3 : 1);
            // A frag (16x4 f32): VGPR0 = K {0|2}, VGPR1 = K {1|3} by lane half.
            v2f a, b0, b1;
            a[0]  = As[buf][wm * 16 + l][klo];
            a[1]  = As[buf][wm * 16 + l][khi];
            // B frags (4x16 f32): column N = l per lane, K split by lane half.
            b0[0] = Bs[buf][wn * 32 + l][klo];
            b0[1] = Bs[buf][wn * 32 + l][khi];
            b1[0] = Bs[buf][wn * 32 + 16 + l][klo];
            b1[1] = Bs[buf][wn * 32 + 16 + l][khi];
            acc0 = __builtin_amdgcn_wmma_f32_16x16x4_f32(
                false, a, false, b0, (short)0, acc0, false, false);
            acc1 = __builtin_amdgcn_wmma_f32_16x16x4_f32(
                false, a, false, b1, (short)0, acc1, false, false);
        }
        __syncthreads();                   // all waves done reading `buf`
    }

    // Store: VGPR r holds M = r (lanes 0-15) / r+8 (lanes 16-31), N = l.
    size_t row_base = bm + (size_t)wm * 16;
    int    col0     = bn + wn * 32 + l;
#pragma unroll
    for (int r = 0; r < 8; ++r) {
        size_t m = row_base + r + 8 * half;
        float v0 = acc0[r];
        float v1 = acc1[r];
        if (R) {
            v0 += R[m * N + col0];
            v1 += R[m * N + col0 + 16];
        }
        C[m * N + col0]      = v0;
        C[m * N + col0 + 16] = v1;
    }
}

// ---------------------------------------------------------------------------
// Depthwise causal (rev=0) / anti-causal (rev=1) conv1d, K=4, + bias + SiLU.
// ---------------------------------------------------------------------------
__global__ __launch_bounds__(256)
void conv_silu_kernel(const float* __restrict__ xz, const float* __restrict__ cw,
                      const float* __restrict__ cb, float* __restrict__ xs,
                      int rev) {
    size_t gid = (size_t)blockIdx.x * 256 + threadIdx.x;
    if (gid >= (size_t)NTOK * D_INNER) return;
    int    d = (int)(gid % D_INNER);
    size_t m = gid / D_INNER;
    int    t = (int)(m % SEQ);
    size_t bb = m / SEQ;

    const float* w = cw + (size_t)d * K_CONV;
    float acc = cb[d];
#pragma unroll
    for (int j = 0; j < K_CONV; ++j) {
        int tt = rev ? (t + j) : (t - j);
        if (tt >= 0 && tt < SEQ) {
            size_t mm = bb * SEQ + tt;
            acc += w[K_CONV - 1 - j] * xz[mm * (2 * D_INNER) + d];
        }
    }
    xs[m * D_INNER + d] = silu_f(acc);
}

// ---------------------------------------------------------------------------
// x_proj: x_dbl[m, o] = sum_k xs[m,k] * W[o,k]   (o < 56, K = 768)
// ---------------------------------------------------------------------------
__global__ __launch_bounds__(64)
void xproj_kernel(const float* __restrict__ xs, const float* __restrict__ W,
                  float* __restrict__ xdbl) {
    int m = blockIdx.x;
    int o = threadIdx.x;
    if (o >= XPROJ_N) return;
    const float* a = xs + (size_t)m * D_INNER;
    const float* w = W + (size_t)o * D_INNER;
    float s = 0.0f;
#pragma unroll 8
    for (int k = 0; k < D_INNER; ++k) s += a[k] * w[k];
    xdbl[(size_t)m * XPROJ_N + o] = s;
}

// ---------------------------------------------------------------------------
// dt[m, d] = softplus( sum_r x_dbl[m,r] * dt_w[d,r] + dt_b[d] ),  r < 24
// ---------------------------------------------------------------------------
__global__ __launch_bounds__(256)
void dt_kernel(const float* __restrict__ xdbl, const float* __restrict__ dtw,
               const float* __restrict__ dtb, float* __restrict__ dt) {
    size_t gid = (size_t)blockIdx.x * 256 + threadIdx.x;
    if (gid >= (size_t)NTOK * D_INNER) return;
    int    d = (int)(gid % D_INNER);
    size_t m = gid / D_INNER;
    const float* r = xdbl + m * XPROJ_N;
    const float* w = dtw + (size_t)d * DT_RANK;
    float s = dtb[d];
#pragma unroll
    for (int i = 0; i < DT_RANK; ++i) s += r[i] * w[i];
    dt[gid] = (s > 20.0f) ? s : log1pf(__expf(s));
}

// ---------------------------------------------------------------------------
// Selective scan + D-skip + SiLU gate. One thread per (batch, channel d):
// h[16], A[16] in registers; shared B_t/C_t staged in LDS (float4 loads +
// b128 stores) per 32-step time chunk. rev=1 scans back-to-front.
// ---------------------------------------------------------------------------
#define SCAN_TT 32
__global__ __launch_bounds__(256)
void scan_kernel(const float* __restrict__ dt, const float* __restrict__ xs,
                 const float* __restrict__ xdbl, const float* __restrict__ xz,
                 const float* __restrict__ A_log, const float* __restrict__ Dp,
                 float* __restrict__ yg, int rev) {
    __shared__ __align__(16) float Bs[SCAN_TT][D_STATE];
    __shared__ __align__(16) float Cs[SCAN_TT][D_STATE];

    int tid = threadIdx.x;
    int d   = blockIdx.x * 256 + tid;   // channel 0..767
    int b   = blockIdx.y;               // batch

    // staging map: 32 steps x 16 floats x {B,C} = 256 float4's, one per thread
    int t2     = tid & 127;
    int sti    = t2 >> 2;               // step within chunk (0..31)
    int sq     = (t2 & 3) * 4;          // float4 column (0,4,8,12)
    int whichC = tid >> 7;              // 0: stage B, 1: stage C

    float A[D_STATE], h[D_STATE];
#pragma unroll
    for (int n = 0; n < D_STATE; ++n) {
        A[n] = -__expf(A_log[(size_t)d * D_STATE + n]);
        h[n] = 0.0f;
    }
    float Dv = Dp[d];

    for (int s0 = 0; s0 < SEQ; s0 += SCAN_TT) {
        __syncthreads();
        {
            int s = s0 + sti;
            int t = rev ? (SEQ - 1 - s) : s;
            size_t row = ((size_t)b * SEQ + t) * XPROJ_N;
            float4 v = *(const float4*)&xdbl[row + DT_RANK + whichC * D_STATE + sq];
            float* dst = whichC ? &Cs[sti][sq] : &Bs[sti][sq];
            *(float4*)dst = v;
        }
        __syncthreads();

        for (int i = 0; i < SCAN_TT; ++i) {
            int s = s0 + i;
            int t = rev ? (SEQ - 1 - s) : s;
            size_t m = (size_t)b * SEQ + t;
            float dtv = dt[m * D_INNER + d];
            float xv  = xs[m * D_INNER + d];
            float dx  = dtv * xv;
            float y   = 0.0f;
#pragma unroll
            for (int n = 0; n < D_STATE; ++n) {
                float da = __expf(dtv * A[n]);
                h[n] = da * h[n] + dx * Bs[i][n];
                y += h[n] * Cs[i][n];
            }
            y += xv * Dv;
            float z = xz[m * (2 * D_INNER) + D_INNER + d];
            yg[m * D_INNER + d] = y * silu_f(z);
        }
    }
}

// ---------------------------------------------------------------------------
// Host-side orchestration
// ---------------------------------------------------------------------------
extern "C" void kernel_launch(void* const* d_in, const int* in_sizes, int n_in,
                              void* d_out, int out_size, void* d_ws, size_t ws_size,
                              hipStream_t stream) {
    (void)in_sizes; (void)n_in; (void)out_size; (void)ws_size;

    const float* x        = (const float*)d_in[0];
    const float* norm_w   = (const float*)d_in[1];
    const float* norm_b   = (const float*)d_in[2];
    const float* in_proj  = (const float*)d_in[3];
    const float* conv_w   = (const float*)d_in[4];
    const float* conv_b   = (const float*)d_in[5];
    const float* x_proj   = (const float*)d_in[6];
    const float* dt_w     = (const float*)d_in[7];
    const float* dt_b     = (const float*)d_in[8];
    const float* A_log    = (const float*)d_in[9];
    const float* Dp       = (const float*)d_in[10];
    const float* out_proj = (const float*)d_in[11];

    float* ws = (float*)d_ws;
    size_t off = 0;
    float* h_buf  = ws + off; off += (size_t)NTOK * D_MODEL;       // residual stream
    float* xn     = ws + off; off += (size_t)NTOK * D_MODEL;       // layernorm out
    float* xz     = ws + off; off += (size_t)NTOK * 2 * D_INNER;   // in_proj out
    float* xs     = ws + off; off += (size_t)NTOK * D_INNER;       // conv+silu out
    float* dt     = ws + off; off += (size_t)NTOK * D_INNER;       // softplus(dt)
    float* xdbl   = ws + off; off += (size_t)NTOK * XPROJ_N;       // x_proj out
    float* yg     = ws + off; off += (size_t)NTOK * D_INNER;       // gated scan out

    for (int dir = 0; dir < 2; ++dir) {
        const float* lin   = dir ? h_buf : x;     // LN input == residual input
        float*       dst   = dir ? (float*)d_out : h_buf;
        const int    rev   = dir;                 // block 1 runs time-reversed

        const float* nw  = norm_w   + (size_t)dir * D_MODEL;
        const float* nb  = norm_b   + (size_t)dir * D_MODEL;
        const float* ipw = in_proj  + (size_t)dir * 2 * D_INNER * D_MODEL;
        const float* cw  = conv_w   + (size_t)dir * D_INNER * K_CONV;
        const float* cb  = conv_b   + (size_t)dir * D_INNER;
        const float* xpw = x_proj   + (size_t)dir * XPROJ_N * D_INNER;
        const float* dtw = dt_w     + (size_t)dir * D_INNER * DT_RANK;
        const float* dtb = dt_b     + (size_t)dir * D_INNER;
        const float* alg = A_log    + (size_t)dir * D_INNER * D_STATE;
        const float* dpp = Dp       + (size_t)dir * D_INNER;
        const float* opw = out_proj + (size_t)dir * D_MODEL * D_INNER;

        // 1. LayerNorm
        ln_kernel<<<NTOK / 4, 128, 0, stream>>>(lin, nw, nb, xn);

        // 2. in_proj: (16384 x 384) x (384 x 1536)
        gemm_wmma_kernel<<<dim3((2 * D_INNER) / 64, NTOK / 64), 256, 0, stream>>>(
            xn, ipw, nullptr, xz, NTOK, 2 * D_INNER, D_MODEL);

        // 3. depthwise conv + SiLU
        conv_silu_kernel<<<((size_t)NTOK * D_INNER) / 256, 256, 0, stream>>>(
            xz, cw, cb, xs, rev);

        // 4. x_proj: (16384 x 768) x (768 x 56)
        xproj_kernel<<<NTOK, 64, 0, stream>>>(xs, xpw, xdbl);

        // 5. dt projection + softplus
        dt_kernel<<<((size_t)NTOK * D_INNER) / 256, 256, 0, stream>>>(
            xdbl, dtw, dtb, dt);

        // 6. selective scan (+D skip, +SiLU gate)
        scan_kernel<<<dim3(D_INNER / 256, BATCH), 256, 0, stream>>>(
            dt, xs, xdbl, xz, alg, dpp, yg, rev);

        // 7. out_proj + residual: (16384 x 768) x (768 x 384) + lin
        gemm_wmma_kernel<<<dim3(D_MODEL / 64, NTOK / 64), 256, 0, stream>>>(
            yg, opw, lin, dst, NTOK, D_MODEL, D_INNER);
    }
}